// FuzzyCNFDiscriminator_53730040873552
// MI455X (gfx1250) — compile-verified
//
#include <hip/hip_runtime.h>
#include <stdint.h>

// ---------------- problem constants (match reference) ----------------
#define NUM_VARS    10000
#define NUM_CLAUSES 100000
#define KLITS       3
#define BATCH       256
#define NVB         (NUM_VARS * BATCH)            // 2,560,000 elements
#define NVB4        (NVB / 4)                     // 640,000 float4s

// tile of clauses staged into LDS per async step
#define TILE        256
#define NTILES      ((NUM_CLAUSES + TILE - 1) / TILE)   // 391
#define PAD_CLAUSES (NTILES * TILE)                     // 100096
#define NBLOCKS_FULL 196
#define NBLOCKS_META 128

#define POS_INF_BITS 0x7F800000u

// ---------------- CDNA5 async helpers ----------------
__device__ __forceinline__ void async_tile_load_b128(const void* gsrc, unsigned lds_byte_addr) {
#if defined(__gfx1250__)
    // GV mode: 64-bit global address in a VGPR pair, LDS byte address in VDST VGPR.
    // Tracked by ASYNCcnt (CDNA5 ISA).
    uint64_t ga = (uint64_t)(uintptr_t)gsrc;
    asm volatile("global_load_async_to_lds_b128 %0, %1, off"
                 :
                 : "v"(lds_byte_addr), "v"(ga)
                 : "memory");
#else
    (void)gsrc; (void)lds_byte_addr;
#endif
}

__device__ __forceinline__ void wait_async0() {
#if defined(__gfx1250__)
#if __has_builtin(__builtin_amdgcn_s_wait_asynccnt)
    __builtin_amdgcn_s_wait_asynccnt(0);
#else
    asm volatile("s_wait_asynccnt 0x0" ::: "memory");
#endif
#endif
}

// ---------------- kernel 0: init output to +inf ----------------
__global__ void fuzzy_init_out(unsigned* __restrict__ out, int n) {
    int i = blockIdx.x * blockDim.x + threadIdx.x;
    if (i < n) out[i] = POS_INF_BITS;
}

// ---------------- kernel 1a: build pos/neg table in workspace ----------------
// pn[0 .. NVB)      = x
// pn[NVB .. 2*NVB)  = 1 - x        (Goedel negation precomputed once)
__global__ void fuzzy_build_posneg(const float4* __restrict__ in4,
                                   float4* __restrict__ pn4) {
    int i = blockIdx.x * blockDim.x + threadIdx.x;
    if (i >= NVB4) return;
    float4 v = in4[i];
    pn4[i] = v;
    float4 nv;
    nv.x = 1.0f - v.x; nv.y = 1.0f - v.y; nv.z = 1.0f - v.z; nv.w = 1.0f - v.w;
    pn4[NVB4 + i] = nv;
}

// ---------------- kernel 1b: pack clause metadata ----------------
// MODE 0 (meta-only path): m = (idx*BATCH) | (neg<<31), scalar-float addressing
// MODE 1 (full path):      m = idx*64 + neg*NVB4, float4 index into posneg table
template <int MODE>
__global__ void fuzzy_pack_meta(const int* __restrict__ lit_idx,
                                const int* __restrict__ lit_neg,
                                uint4* __restrict__ meta) {
    int c = blockIdx.x * blockDim.x + threadIdx.x;
    if (c >= PAD_CLAUSES) return;
    uint4 m;
    if (c < NUM_CLAUSES) {
        unsigned i0 = (unsigned)lit_idx[c * KLITS + 0];
        unsigned i1 = (unsigned)lit_idx[c * KLITS + 1];
        unsigned i2 = (unsigned)lit_idx[c * KLITS + 2];
        unsigned n0 = (unsigned)lit_neg[c * KLITS + 0];
        unsigned n1 = (unsigned)lit_neg[c * KLITS + 1];
        unsigned n2 = (unsigned)lit_neg[c * KLITS + 2];
        if (MODE == 0) {
            m.x = (i0 * BATCH) | (n0 << 31);
            m.y = (i1 * BATCH) | (n1 << 31);
            m.z = (i2 * BATCH) | (n2 << 31);
        } else {
            m.x = i0 * (BATCH / 4) + n0 * NVB4;
            m.y = i1 * (BATCH / 4) + n1 * NVB4;
            m.z = i2 * (BATCH / 4) + n2 * NVB4;
        }
        m.w = 0u;
    } else {
        m.x = 0u; m.y = 0u; m.z = 0u; m.w = 1u;   // padding clause (skipped via jmax)
    }
    meta[c] = m;
}

// ---------------- kernel 2 (full): float4 gathers from posneg table ----------------
// Block = 256 threads = 4 clause-subgroups x 64 column-groups (float4 per thread).
// Each wave has a uniform clause -> LDS metadata read is a broadcast; 32 lanes x 16B
// contiguous per gather -> fully coalesced 512B requests into the L2-resident table.
__global__ __launch_bounds__(256)
void fuzzy_cnf_full(const float4* __restrict__ pn4,   // [2*NVB4] pos/neg table
                    const uint4* __restrict__ meta,   // [PAD_CLAUSES]
                    unsigned* __restrict__ out) {     // [BATCH] as f32 bits
    __shared__ uint4 buf[2][TILE];                    // 8 KB double buffer

    const int tid = threadIdx.x;
    const int q   = tid >> 6;                         // clause subgroup 0..3 (wave-uniform)
    const int c   = tid & 63;                         // float4 column group 0..63
    const int nb  = gridDim.x;

    float4 acc;
    acc.x = acc.y = acc.z = acc.w = __int_as_float((int)POS_INF_BITS);

    // prologue: stage first tile (256 lanes x b128 = 4KB)
    int tile0 = blockIdx.x;
    if (tile0 < NTILES) {
        unsigned ldsa = (unsigned)(uintptr_t)&buf[0][tid];
        async_tile_load_b128(&meta[(size_t)tile0 * TILE + tid], ldsa);
    }

    int cur = 0;
    for (int tl = tile0; tl < NTILES; tl += nb) {
        wait_async0();
        __syncthreads();

        int tn = tl + nb;
        if (tn < NTILES) {
            unsigned ldsa = (unsigned)(uintptr_t)&buf[cur ^ 1][tid];
            async_tile_load_b128(&meta[(size_t)tn * TILE + tid], ldsa);
        }

        const int jmax = NUM_CLAUSES - tl * TILE;      // 256, or 160 on the last tile
        #pragma unroll 2
        for (int jj = 0; jj < TILE / 4; ++jj) {
            int j = 4 * jj + q;
            if (j >= jmax) break;                      // wave-uniform branch
            uint4 m = buf[cur][j];                     // LDS broadcast within wave
            float4 x0 = pn4[m.x + (unsigned)c];        // negation baked into address
            float4 x1 = pn4[m.y + (unsigned)c];
            float4 x2 = pn4[m.z + (unsigned)c];
            float4 cv;
            cv.x = fmaxf(fmaxf(x0.x, x1.x), x2.x);
            cv.y = fmaxf(fmaxf(x0.y, x1.y), x2.y);
            cv.z = fmaxf(fmaxf(x0.z, x1.z), x2.z);
            cv.w = fmaxf(fmaxf(x0.w, x1.w), x2.w);
            acc.x = fminf(acc.x, cv.x);
            acc.y = fminf(acc.y, cv.y);
            acc.z = fminf(acc.z, cv.z);
            acc.w = fminf(acc.w, cv.w);
        }

        __syncthreads();
        cur ^= 1;
    }

    // cross-block min: all values in [0,1] (non-negative) -> IEEE order == u32 order.
    atomicMin(&out[4 * c + 0], __float_as_uint(acc.x));
    atomicMin(&out[4 * c + 1], __float_as_uint(acc.y));
    atomicMin(&out[4 * c + 2], __float_as_uint(acc.z));
    atomicMin(&out[4 * c + 3], __float_as_uint(acc.w));
}

// ---------------- kernel 2 (meta-only fallback): scalar gathers ----------------
__global__ __launch_bounds__(BATCH)
void fuzzy_cnf_meta(const float* __restrict__ input,
                    const uint4* __restrict__ meta,
                    unsigned* __restrict__ out) {
    __shared__ uint4 buf[2][TILE];
    const int t  = threadIdx.x;
    const int nb = gridDim.x;
    float acc = __int_as_float((int)POS_INF_BITS);

    int tile0 = blockIdx.x;
    if (tile0 < NTILES) {
        unsigned ldsa = (unsigned)(uintptr_t)&buf[0][t];
        async_tile_load_b128(&meta[(size_t)tile0 * TILE + t], ldsa);
    }
    int cur = 0;
    for (int tl = tile0; tl < NTILES; tl += nb) {
        wait_async0();
        __syncthreads();
        int tn = tl + nb;
        if (tn < NTILES) {
            unsigned ldsa = (unsigned)(uintptr_t)&buf[cur ^ 1][t];
            async_tile_load_b128(&meta[(size_t)tn * TILE + t], ldsa);
        }
        const int jmax = (NUM_CLAUSES - tl * TILE) < TILE ? (NUM_CLAUSES - tl * TILE) : TILE;
        #pragma unroll 4
        for (int j = 0; j < jmax; ++j) {
            uint4 m = buf[cur][j];
            float x0 = input[(m.x & 0x7FFFFFFFu) + (unsigned)t];
            float x1 = input[(m.y & 0x7FFFFFFFu) + (unsigned)t];
            float x2 = input[(m.z & 0x7FFFFFFFu) + (unsigned)t];
            float y0 = fabsf((float)(m.x >> 31) - x0);   // neg? 1-x : x
            float y1 = fabsf((float)(m.y >> 31) - x1);
            float y2 = fabsf((float)(m.z >> 31) - x2);
            acc = fminf(acc, fmaxf(fmaxf(y0, y1), y2));
        }
        __syncthreads();
        cur ^= 1;
    }
    atomicMin(&out[t], __float_as_uint(acc));
}

// ---------------- fallback (no workspace): direct metadata reads ----------------
__global__ __launch_bounds__(BATCH)
void fuzzy_cnf_fallback(const float* __restrict__ input,
                        const int* __restrict__ lit_idx,
                        const int* __restrict__ lit_neg,
                        unsigned* __restrict__ out) {
    const int t = threadIdx.x;
    float acc = __int_as_float((int)POS_INF_BITS);
    for (int c = blockIdx.x; c < NUM_CLAUSES; c += gridDim.x) {
        int i0 = lit_idx[c * KLITS + 0], i1 = lit_idx[c * KLITS + 1], i2 = lit_idx[c * KLITS + 2];
        int n0 = lit_neg[c * KLITS + 0], n1 = lit_neg[c * KLITS + 1], n2 = lit_neg[c * KLITS + 2];
        float x0 = input[(size_t)i0 * BATCH + t];
        float x1 = input[(size_t)i1 * BATCH + t];
        float x2 = input[(size_t)i2 * BATCH + t];
        float y0 = fabsf((float)n0 - x0);
        float y1 = fabsf((float)n1 - x1);
        float y2 = fabsf((float)n2 - x2);
        acc = fminf(acc, fmaxf(fmaxf(y0, y1), y2));
    }
    atomicMin(&out[t], __float_as_uint(acc));
}

// ---------------- launcher ----------------
extern "C" void kernel_launch(void* const* d_in, const int* in_sizes, int n_in,
                              void* d_out, int out_size, void* d_ws, size_t ws_size,
                              hipStream_t stream) {
    const float* input   = (const float*)d_in[0];   // [NUM_VARS, BATCH] f32
    const int*   lit_idx = (const int*)d_in[1];     // [NUM_CLAUSES, 3] int
    const int*   lit_neg = (const int*)d_in[2];     // [NUM_CLAUSES, 3] int
    unsigned*    out     = (unsigned*)d_out;        // [BATCH] f32 bits

    (void)in_sizes; (void)n_in;

    fuzzy_init_out<<<(out_size + 255) / 256, 256, 0, stream>>>(out, out_size);

    const size_t posneg_bytes = (size_t)2 * NVB * sizeof(float);        // 20,480,000
    const size_t meta_bytes   = (size_t)PAD_CLAUSES * sizeof(uint4);    // 1,601,536
    const size_t need_full    = posneg_bytes + meta_bytes;              // ~22.1 MB

    if (ws_size >= need_full && d_ws != nullptr) {
        float4* pn4  = (float4*)d_ws;
        uint4*  meta = (uint4*)((char*)d_ws + posneg_bytes);            // 16B aligned
        fuzzy_build_posneg<<<(NVB4 + 255) / 256, 256, 0, stream>>>((const float4*)input, pn4);
        fuzzy_pack_meta<1><<<(PAD_CLAUSES + 255) / 256, 256, 0, stream>>>(lit_idx, lit_neg, meta);
        fuzzy_cnf_full<<<NBLOCKS_FULL, 256, 0, stream>>>((const float4*)pn4, meta, out);
    } else if (ws_size >= meta_bytes && d_ws != nullptr) {
        uint4* meta = (uint4*)d_ws;
        fuzzy_pack_meta<0><<<(PAD_CLAUSES + 255) / 256, 256, 0, stream>>>(lit_idx, lit_neg, meta);
        fuzzy_cnf_meta<<<NBLOCKS_META, BATCH, 0, stream>>>(input, meta, out);
    } else {
        fuzzy_cnf_fallback<<<256, BATCH, 0, stream>>>(input, lit_idx, lit_neg, out);
    }
}